// AttentionWithLoc_61959198212312
// MI455X (gfx1250) — compile-verified
//
#include <hip/hip_runtime.h>
#include <math.h>

typedef __attribute__((ext_vector_type(2))) float v2f;
typedef __attribute__((ext_vector_type(8))) float v8f;

#define B_    16
#define C_    684
#define H_    32
#define W_    128
#define HW_   4096          // H_*W_
#define HID_  256
#define A_    512
#define COV_  512
#define KS_   11
#define PAD_  5
#define TAPS_ 121
#define KPAD_ 124           // taps padded to multiple of 4 for wmma K-loop
#define TROW_ 128           // padded tap rows (64 K-pairs)

// ---- workspace layout (floats) ----
#define WS_QUERY  0                         // B_*A_      = 8192
#define WS_WCT    (WS_QUERY + B_*A_)        // TROW_*A_   = 65536 (paired layout)
#define WS_ENERGY (WS_WCT + TROW_*A_)       // B_*HW_     = 65536
#define WS_BMAX   (WS_ENERGY + B_*HW_)      // 4096
#define WS_GMAX   (WS_BMAX + 4096)          // 1

// ---------------------------------------------------------------------------
// K1a: query[b,a] = hidden[b,:] . W_hidden[a,:] + b_hidden[a]
// ---------------------------------------------------------------------------
__global__ __launch_bounds__(256) void query_kernel(
    const float* __restrict__ hidden, const float* __restrict__ Wh,
    const float* __restrict__ bh, float* __restrict__ query) {
  int gid = blockIdx.x * 256 + threadIdx.x;
  if (gid >= B_ * A_) return;
  int b = gid >> 9;          // / A_
  int a = gid & (A_ - 1);
  const float* hr = hidden + b * HID_;
  const float* wr = Wh + a * HID_;
  float s = bh[a];
  for (int k = 0; k < HID_; ++k) s = fmaf(hr[k], wr[k], s);
  query[gid] = s;
}

// ---------------------------------------------------------------------------
// K1b: W_combined in K-paired layout:
//   WcP[t>>1][a][t&1] = sum_k W_attn[a,k] * W_attconv[k,0,t]   (t<121, else 0)
// so a lane's B fragment (K=k, k+1 at channel a) is one contiguous float2.
// ---------------------------------------------------------------------------
__global__ __launch_bounds__(256) void wct_kernel(
    const float* __restrict__ Wattn, const float* __restrict__ Wconv,
    float* __restrict__ WcP) {
  int gid = blockIdx.x * 256 + threadIdx.x;   // gid = t*A_ + a
  if (gid >= TROW_ * A_) return;
  int t = gid >> 9;
  int a = gid & (A_ - 1);
  float s = 0.f;
  if (t < TAPS_) {
    const float* wr = Wattn + a * COV_;
    for (int k = 0; k < COV_; ++k) s = fmaf(wr[k], Wconv[k * TAPS_ + t], s);
  }
  WcP[(((t >> 1) * A_) + a) * 2 + (t & 1)] = s;
}

// ---------------------------------------------------------------------------
// K2: fused coverage-conv (WMMA f32 GEMM) + tanh + energy dot + block max
// One block = 16 consecutive pixels in one image row; 8 waves x 64 channels.
// ---------------------------------------------------------------------------
__global__ __launch_bounds__(256) void fused_energy_kernel(
    const float* __restrict__ alpha_sum,   // [B,1,H,W]
    const float* __restrict__ cnn_trans,   // [B,A,H,W]
    const float* __restrict__ counting,    // [B,H,W,A]
    const float* __restrict__ query,       // [B,A]
    const float* __restrict__ WcP,         // [TROW_/2][A_][2] K-paired
    const float* __restrict__ W_alpha,     // [A]
    const float* __restrict__ b_alpha,     // [1]
    float* __restrict__ energy,            // [B,H,W]
    float* __restrict__ blockmax) {        // [gridDim.x]
  __shared__ float patch[16 * 129];        // 129-pitch: conflict-free
  __shared__ float e_acc[16];

  const int tid = threadIdx.x;
  const int g = blockIdx.x;                // 0..4095
  const int b = g >> 8;                    // / (HW_/16)
  const int pt = g & 255;
  const int h = pt >> 3;                   // 8 tiles per 128-wide row
  const int w0 = (pt & 7) << 4;
  const int pbase = h * W_ + w0;

  // im2col: patch[m][t] = alpha_sum[b, h+di-5, w0+m+dj-5], zero padded
  for (int idx = tid; idx < 16 * 128; idx += 256) {
    int m = idx >> 7;
    int t = idx & 127;
    float v = 0.f;
    if (t < TAPS_) {
      int di = t / KS_, dj = t - di * KS_;
      int hh = h + di - PAD_;
      int ww = w0 + m + dj - PAD_;
      if (hh >= 0 && hh < H_ && ww >= 0 && ww < W_)
        v = alpha_sum[(b * H_ + hh) * W_ + ww];
    }
    patch[m * 129 + t] = v;
  }
  if (tid < 16) e_acc[tid] = 0.f;
  __syncthreads();

  const int wave = tid >> 5;
  const int lane = tid & 31;
  const int half = lane >> 4;
  const int l16 = lane & 15;

  // B fragments: one float2 per lane per tile per K-step, fully coalesced
  const float2* wrow = reinterpret_cast<const float2*>(WcP);
  const int nbase = wave * 64 + l16;       // + t*16 per tile

  v8f c[4] = {};                           // 4 channel tiles of 16
#pragma unroll
  for (int k0 = 0; k0 < KPAD_; k0 += 4) {
    v2f af;                                // A: patch 16x4, M=l16, K by half
    af.x = patch[l16 * 129 + k0 + 2 * half];
    af.y = patch[l16 * 129 + k0 + 2 * half + 1];
    const int krow = ((k0 >> 1) + half) * A_ + nbase;
#pragma unroll
    for (int t = 0; t < 4; ++t) {
      float2 w = wrow[krow + t * 16];      // single global_load_b64
      v2f bf;
      bf.x = w.x;
      bf.y = w.y;
      c[t] = __builtin_amdgcn_wmma_f32_16x16x4_f32(
          false, af, false, bf, (short)0, c[t], false, false);
    }
  }

  // epilogue: + query + cnn_trans + counting, tanh, * W_alpha, reduce
  float q[4], wa[4];
  int ch[4];
  float tr[4][8];
#pragma unroll
  for (int t = 0; t < 4; ++t) {
    ch[t] = wave * 64 + t * 16 + l16;
    wa[t] = W_alpha[ch[t]];
    q[t] = query[b * A_ + ch[t]];
    const float4* cp = reinterpret_cast<const float4*>(
        cnn_trans + (b * A_ + ch[t]) * HW_ + pbase + 8 * half);
    float4 v0 = cp[0], v1 = cp[1];
    tr[t][0] = v0.x; tr[t][1] = v0.y; tr[t][2] = v0.z; tr[t][3] = v0.w;
    tr[t][4] = v1.x; tr[t][5] = v1.y; tr[t][6] = v1.z; tr[t][7] = v1.w;
  }

#pragma unroll
  for (int r = 0; r < 8; ++r) {
    int p = pbase + r + 8 * half;          // pixel of C row r for this lane
    float acc = 0.f;
#pragma unroll
    for (int t = 0; t < 4; ++t) {
      float cnt = counting[(b * HW_ + p) * A_ + ch[t]];
      float x = c[t][r] + q[t] + cnt + tr[t][r];
      acc += wa[t] * tanhf(x);
    }
    // reduce over the 16 lanes of each half (channels) -> per-pixel partial
    acc += __shfl_xor(acc, 1, 32);
    acc += __shfl_xor(acc, 2, 32);
    acc += __shfl_xor(acc, 4, 32);
    acc += __shfl_xor(acc, 8, 32);
    if (l16 == 0) atomicAdd(&e_acc[r + 8 * half], acc);
  }
  __syncthreads();

  const float ba = b_alpha[0];
  if (tid < 16) energy[b * HW_ + pbase + tid] = e_acc[tid] + ba;
  if (tid == 0) {
    float mx = -INFINITY;
    for (int i = 0; i < 16; ++i) mx = fmaxf(mx, e_acc[i] + ba);
    blockmax[blockIdx.x] = mx;
  }
}

// ---------------------------------------------------------------------------
// K3: global max of 4096 block maxima
// ---------------------------------------------------------------------------
__global__ __launch_bounds__(256) void gmax_kernel(
    const float* __restrict__ bmax, float* __restrict__ gmax) {
  __shared__ float red[256];
  float m = -INFINITY;
  for (int i = threadIdx.x; i < 4096; i += 256) m = fmaxf(m, bmax[i]);
  red[threadIdx.x] = m;
  __syncthreads();
  for (int s = 128; s > 0; s >>= 1) {
    if (threadIdx.x < s)
      red[threadIdx.x] = fmaxf(red[threadIdx.x], red[threadIdx.x + s]);
    __syncthreads();
  }
  if (threadIdx.x == 0) gmax[0] = red[0];
}

// ---------------------------------------------------------------------------
// K4: per-batch masked softmax -> alpha, alpha_sum_new (written to d_out)
// ---------------------------------------------------------------------------
__global__ __launch_bounds__(256) void softmax_kernel(
    const float* __restrict__ energy, const int* __restrict__ mask,
    const float* __restrict__ alpha_sum, const float* __restrict__ gmax,
    float* __restrict__ out_alpha, float* __restrict__ out_asum) {
  __shared__ float red[256];
  __shared__ float inv;
  const int b = blockIdx.x;
  const float gm = gmax[0];
  const float* e = energy + b * HW_;
  const int* mk = mask + b * HW_;
  float s = 0.f;
  for (int i = threadIdx.x; i < HW_; i += 256)
    if (mk[i]) s += __expf(e[i] - gm);
  red[threadIdx.x] = s;
  __syncthreads();
  for (int st = 128; st > 0; st >>= 1) {
    if (threadIdx.x < st) red[threadIdx.x] += red[threadIdx.x + st];
    __syncthreads();
  }
  if (threadIdx.x == 0) inv = 1.0f / red[0];
  __syncthreads();
  for (int i = threadIdx.x; i < HW_; i += 256) {
    float al = mk[i] ? __expf(e[i] - gm) * inv : 0.f;
    out_alpha[b * HW_ + i] = al;
    out_asum[b * HW_ + i] = al + alpha_sum[b * HW_ + i];
  }
}

// ---------------------------------------------------------------------------
// K5: context[b,c] = sum_p alpha[b,p] * cnn[b,c,p]  (memory bound, coalesced)
// ---------------------------------------------------------------------------
__global__ __launch_bounds__(256) void context_kernel(
    const float* __restrict__ cnn, const float* __restrict__ alpha,
    float* __restrict__ ctx) {
  __shared__ float red[256];
  const int bc = blockIdx.x;                 // b*C_ + c
  const int b = bc / C_;
  const float* f = cnn + (size_t)bc * HW_;
  const float* al = alpha + b * HW_;
  float s = 0.f;
  for (int i = threadIdx.x; i < HW_; i += 256) {
    __builtin_prefetch(f + i + 1024, 0, 0);  // global_prefetch_b8
    s = fmaf(f[i], al[i], s);
  }
  red[threadIdx.x] = s;
  __syncthreads();
  for (int st = 128; st > 0; st >>= 1) {
    if (threadIdx.x < st) red[threadIdx.x] += red[threadIdx.x + st];
    __syncthreads();
  }
  if (threadIdx.x == 0) ctx[bc] = red[0];
}

// ---------------------------------------------------------------------------
extern "C" void kernel_launch(void* const* d_in, const int* in_sizes, int n_in,
                              void* d_out, int out_size, void* d_ws,
                              size_t ws_size, hipStream_t stream) {
  const float* cnn_features = (const float*)d_in[0];   // [B,C,H,W]
  const float* cnn_trans    = (const float*)d_in[1];   // [B,A,H,W]
  const float* hidden       = (const float*)d_in[2];   // [B,HID]
  const float* alpha_sum    = (const float*)d_in[3];   // [B,1,H,W]
  const int*   image_mask   = (const int*)d_in[4];     // [B,1,H,W]
  const float* counting     = (const float*)d_in[5];   // [B,H,W,A]
  const float* W_hidden     = (const float*)d_in[6];   // [A,HID]
  const float* b_hidden     = (const float*)d_in[7];   // [A]
  const float* W_attconv    = (const float*)d_in[8];   // [COV,1,11,11]
  const float* W_attn       = (const float*)d_in[9];   // [A,COV]
  const float* W_alpha      = (const float*)d_in[10];  // [1,A]
  const float* b_alpha      = (const float*)d_in[11];  // [1]

  float* ws = (float*)d_ws;
  float* ws_query  = ws + WS_QUERY;
  float* ws_wct    = ws + WS_WCT;
  float* ws_energy = ws + WS_ENERGY;
  float* ws_bmax   = ws + WS_BMAX;
  float* ws_gmax   = ws + WS_GMAX;

  float* out_ctx   = (float*)d_out;                    // [B*C]
  float* out_alpha = out_ctx + B_ * C_;                // [B*HW]
  float* out_asum  = out_alpha + B_ * HW_;             // [B*HW]

  query_kernel<<<(B_ * A_ + 255) / 256, 256, 0, stream>>>(
      hidden, W_hidden, b_hidden, ws_query);
  wct_kernel<<<(TROW_ * A_ + 255) / 256, 256, 0, stream>>>(
      W_attn, W_attconv, ws_wct);
  fused_energy_kernel<<<B_ * HW_ / 16, 256, 0, stream>>>(
      alpha_sum, cnn_trans, counting, ws_query, ws_wct, W_alpha, b_alpha,
      ws_energy, ws_bmax);
  gmax_kernel<<<1, 256, 0, stream>>>(ws_bmax, ws_gmax);
  softmax_kernel<<<B_, 256, 0, stream>>>(
      ws_energy, image_mask, alpha_sum, ws_gmax, out_alpha, out_asum);
  context_kernel<<<B_ * C_, 256, 0, stream>>>(cnn_features, out_alpha, out_ctx);
}